// DAS_PW_46170898432507
// MI455X (gfx1250) — compile-verified
//
#include <hip/hip_runtime.h>
#include <cstdint>
#include <cstddef>

// ---------------- problem constants (match reference) ----------------
#define NANG   4
#define NELEM  128
#define NZ     192
#define NX     192
#define NPIX   (NZ * NX)
#define NSAMP  2048

#define FS_F      20832000.0f
#define C_F       1540.0f
#define FDEMOD_F  5208000.0f
#define FNUM_F    2.0f
#define MINW_F    0.001f
#define INV_2PI_F 0.15915494309189535f

#define BLOCK  256                 // 8 waves (wave32) per workgroup
#define NROWS  (NANG * NELEM)      // 512 staged (angle,element) rows
#define NBUF   4                   // 4-deep ring: prefetch distance 2, 1 barrier/iter

typedef int v4i __attribute__((ext_vector_type(4)));
typedef __attribute__((address_space(1))) v4i*  as1_v4i;
typedef __attribute__((address_space(3))) void* as3_void;
typedef __attribute__((address_space(3))) v4i*  as3_v4i;

// ---------------- CDNA5 async global->LDS copy path ----------------
#if __has_builtin(__builtin_amdgcn_global_load_async_to_lds_b128)
#define HAVE_ASYNC_BUILTIN 1
#else
#define HAVE_ASYNC_BUILTIN 0
#endif

#if __has_builtin(__builtin_amdgcn_s_wait_asynccnt)
#define WAIT_ASYNC_N(n) __builtin_amdgcn_s_wait_asynccnt(n)
#else
#define WAIT_ASYNC_N(n) asm volatile("s_wait_asynccnt " #n ::: "memory")
#endif

// Hardware trig on "turns" (theta / 2pi): v_sin_f32 / v_cos_f32.
#if __has_builtin(__builtin_amdgcn_sinf)
__device__ __forceinline__ float hw_sin_turns(float t) { return __builtin_amdgcn_sinf(t); }
__device__ __forceinline__ float hw_cos_turns(float t) { return __builtin_amdgcn_cosf(t); }
#else
__device__ __forceinline__ float hw_sin_turns(float t) { return sinf(6.2831853071795865f * t); }
__device__ __forceinline__ float hw_cos_turns(float t) { return cosf(6.2831853071795865f * t); }
#endif

// Copy 16 bytes from global memory to LDS asynchronously (ASYNCcnt-tracked).
__device__ __forceinline__ void async_copy16(const float* g, float* l) {
#if defined(__HIP_DEVICE_COMPILE__)
#if HAVE_ASYNC_BUILTIN
  as1_v4i  g4 = (as1_v4i)(uintptr_t)g;   // inttoptr into AS1 (global)
  as3_void lv = (as3_void)l;             // addrspacecast generic -> AS3
  as3_v4i  l4 = (as3_v4i)lv;             // reinterpret pointee within AS3
  __builtin_amdgcn_global_load_async_to_lds_b128(g4, l4, /*offset=*/0, /*cpol=*/0);
#else
  unsigned lds_off = (unsigned)(uintptr_t)(as3_void)l;
  asm volatile("global_load_async_to_lds_b128 %0, %1, off"
               :: "v"(lds_off), "v"(g)
               : "memory");
#endif
#else
  (void)g; (void)l;
#endif
}

// Stage one 2048-sample I row and one Q row (16 KB) into LDS.
// 256 threads * 2 float4 per row => 4 async b128 issues per thread/wave.
__device__ __forceinline__ void stage_row(float* dI, float* dQ,
                                          const float* gI, const float* gQ,
                                          int tid) {
#pragma unroll
  for (int j = 0; j < NSAMP / (4 * BLOCK); ++j) {   // 2 iterations
    const int f4 = tid + j * BLOCK;                 // float4 index 0..511
    async_copy16(gI + 4 * f4, dI + 4 * f4);
    async_copy16(gQ + 4 * f4, dQ + 4 * f4);
  }
}

__device__ __forceinline__ float sel4(float v0, float v1, float v2, float v3, int a) {
  float r = v0;
  r = (a == 1) ? v1 : r;
  r = (a == 2) ? v2 : r;
  r = (a == 3) ? v3 : r;
  return r;
}

// ---------------- main beamforming kernel ----------------
__global__ __launch_bounds__(BLOCK)
void das_pw_kernel(const float* __restrict__ idata,
                   const float* __restrict__ qdata,
                   const float* __restrict__ grid,
                   const float* __restrict__ angles,
                   const float* __restrict__ ele_pos,
                   const float* __restrict__ time_zero,
                   float* __restrict__ out) {
  __shared__ float sI[NBUF][NSAMP];   // 32 KB
  __shared__ float sQ[NBUF][NSAMP];   // 32 KB

  const int tid = threadIdx.x;
  const int p   = blockIdx.x * BLOCK + tid;   // one pixel per thread

  const float x = grid[3 * p + 0];
  const float z = grid[3 * p + 2];
  const float fs_c  = FS_F / C_F;
  const float xlim0 = ele_pos[0];
  const float xlim1 = ele_pos[3 * (NELEM - 1)];

  // Per-angle precompute (4 angles), kept in registers, cndmask-selected.
  // |angle| <= 0.16 rad, so hardware turns-trig is exact to well below the
  // fp32 error already inherent in the reference's delay math. This keeps
  // ocml (and its scalarized Payne-Hanek slow path) out of the kernel.
  float txdelA[NANG];
  float txapoA[NANG];
#pragma unroll
  for (int a = 0; a < NANG; ++a) {
    const float ang = angles[a];
    const float t0  = time_zero[a];
    const float at  = ang * INV_2PI_F;          // radians -> turns
    const float sa  = hw_sin_turns(at);
    const float ca  = hw_cos_turns(at);
    const float ta  = sa / ca;                  // tan(ang)
    txdelA[a] = (x * sa + z * ca + t0 * C_F) * fs_c;
    const float xproj = x - z * ta;
    txapoA[a] = (xproj >= xlim0 * 1.2f && xproj <= xlim1 * 1.2f) ? 1.0f : 0.0f;
  }

  // Phase in revolutions: theta/(2*pi) = (FDEMOD/FS)*delay - (2*FDEMOD/C)*z.
  // FDEMOD/FS == 0.25 exactly for these constants.
  const float rev_per_samp = FDEMOD_F / FS_F;           // 0.25f
  const float z_rev        = (2.0f * FDEMOD_F / C_F) * z;

  float accI = 0.0f;
  float accQ = 0.0f;

  // Prologue: stage rows 0 and 1 (8 async issues per wave outstanding).
  stage_row(&sI[0][0], &sQ[0][0], idata, qdata, tid);
  stage_row(&sI[1][0], &sQ[1][0], idata + NSAMP, qdata + NSAMP, tid);

  for (int k = 0; k < NROWS; ++k) {
    const int cur = k & (NBUF - 1);
    if (k + 2 < NROWS) {
      // Issue row k+2, then wait so rows k+1,k+2 (8 issues) stay in flight
      // while row k (in-order completion) is guaranteed resident.
      stage_row(&sI[(k + 2) & (NBUF - 1)][0], &sQ[(k + 2) & (NBUF - 1)][0],
                idata + (size_t)(k + 2) * NSAMP,
                qdata + (size_t)(k + 2) * NSAMP, tid);
      WAIT_ASYNC_N(8);
    } else if (k + 1 < NROWS) {
      WAIT_ASYNC_N(4);
    } else {
      WAIT_ASYNC_N(0);
    }
    // Single barrier per iteration: also protects buffer reuse, since the
    // write into buf[(k+2)&3] above happens only after this wave passed the
    // barrier of iteration k-1, i.e. after ALL waves finished reading that
    // buffer at iteration k-2.
    __syncthreads();

    const int a = k >> 7;          // k / NELEM
    const int e = k & (NELEM - 1); // k % NELEM

    const float txdel = sel4(txdelA[0], txdelA[1], txdelA[2], txdelA[3], a);
    const float txapo = sel4(txapoA[0], txapoA[1], txapoA[2], txapoA[3], a);

    const float ex = ele_pos[3 * e];
    const float vx = x - ex;
    const float rxdel = sqrtf(vx * vx + z * z) * fs_c;

    bool keep = fabsf(z) > FNUM_F * fabsf(vx);          // |vz/vx| > FNUM
    keep = keep || (fabsf(vx) <= MINW_F);
    keep = keep || ((vx >=  MINW_F) && (x <= xlim0));
    keep = keep || ((vx <= -MINW_F) && (x >= xlim1));
    const float apod = (keep ? 1.0f : 0.0f) * txapo;

    const float delay = txdel + rxdel;
    const float fi = floorf(delay);
    const float fr = delay - fi;
    const int i0 = (int)fi;
    const int i1 = i0 + 1;

    float iv0 = 0.0f, iv1 = 0.0f, qv0 = 0.0f, qv1 = 0.0f;
    if (i0 >= 0 && i0 < NSAMP) { iv0 = sI[cur][i0]; qv0 = sQ[cur][i0]; }
    if (i1 >= 0 && i1 < NSAMP) { iv1 = sI[cur][i1]; qv1 = sQ[cur][i1]; }

    const float ifoc = (1.0f - fr) * iv0 + fr * iv1;
    const float qfoc = (1.0f - fr) * qv0 + fr * qv1;

    // Phase rotation via hardware trig in "turns" with exact rintf reduction.
    const float rev  = rev_per_samp * delay - z_rev;    // theta / (2*pi)
    const float turn = rev - rintf(rev);                // [-0.5, 0.5]
    const float st = hw_sin_turns(turn);
    const float ct = hw_cos_turns(turn);

    accI += (ifoc * ct - qfoc * st) * apod;
    accQ += (qfoc * ct + ifoc * st) * apod;
  }

  // Single writer per pixel: deterministic, no atomics, no init needed.
  out[p]        = accI;
  out[NPIX + p] = accQ;
}

// ---------------- launcher ----------------
extern "C" void kernel_launch(void* const* d_in, const int* in_sizes, int n_in,
                              void* d_out, int out_size, void* d_ws, size_t ws_size,
                              hipStream_t stream) {
  (void)in_sizes; (void)n_in; (void)out_size; (void)d_ws; (void)ws_size;
  const float* idata     = (const float*)d_in[0];
  const float* qdata     = (const float*)d_in[1];
  const float* grid      = (const float*)d_in[2];
  const float* angles    = (const float*)d_in[3];
  const float* ele_pos   = (const float*)d_in[4];
  const float* time_zero = (const float*)d_in[5];
  float* out = (float*)d_out;

  dim3 g(NPIX / BLOCK);   // 144 blocks x 256 threads, one pixel per thread
  das_pw_kernel<<<g, BLOCK, 0, stream>>>(idata, qdata, grid, angles, ele_pos,
                                         time_zero, out);
}